// NXROGraphPyGModel_10127532884094
// MI455X (gfx1250) — compile-verified
//
#include <hip/hip_runtime.h>

typedef __attribute__((ext_vector_type(2))) float v2f;
typedef __attribute__((ext_vector_type(8))) float v8f;

#define NVARS 2048
#define NEDGE 32768
#define NBATCH 64
#define NHID 16
#define NBASIS 5
#define KTOT (NBASIS * NVARS)          /* 10240 */
#define KCHUNK 512
#define NCHUNK (KTOT / KCHUNK)         /* 20 */
#define CHUNKS_PER_BASIS (NVARS / KCHUNK) /* 4 */
#define UTILES (NVARS / 16)            /* 128 */

// ---------------- small prep kernels ----------------

__global__ void k_emb(const float* __restrict__ t_years,
                      const float* __restrict__ alpha_w,
                      float* __restrict__ embT,   // [5][64]
                      float* __restrict__ alpha)  // [64]
{
    int b = threadIdx.x;
    if (b >= NBATCH) return;
    float t = t_years[b];
    float e0 = 1.0f;
    float e1 = cosf(6.2831853071795864f * t);
    float e2 = sinf(6.2831853071795864f * t);
    float e3 = cosf(12.566370614359172f * t);
    float e4 = sinf(12.566370614359172f * t);
    embT[0 * NBATCH + b] = e0;
    embT[1 * NBATCH + b] = e1;
    embT[2 * NBATCH + b] = e2;
    embT[3 * NBATCH + b] = e3;
    embT[4 * NBATCH + b] = e4;
    float dot = e0 * alpha_w[0] + e1 * alpha_w[1] + e2 * alpha_w[2] +
                e3 * alpha_w[3] + e4 * alpha_w[4];
    alpha[b] = 1.0f / (1.0f + expf(-dot));
}

__global__ void k_deg_init(float* __restrict__ deg) {
    int i = blockIdx.x * blockDim.x + threadIdx.x;
    if (i < NVARS) deg[i] = 1.0f;   // self loop
}

__global__ void k_deg_acc(const int* __restrict__ ei, float* __restrict__ deg) {
    int e = blockIdx.x * blockDim.x + threadIdx.x;
    if (e < NEDGE) atomicAdd(&deg[ei[NEDGE + e]], 1.0f);  // col = ei[1][e]
}

__global__ void k_dinv(const float* __restrict__ deg, float* __restrict__ dinv) {
    int i = blockIdx.x * blockDim.x + threadIdx.x;
    if (i < NVARS) dinv[i] = rsqrtf(deg[i]);
}

__global__ void k_transpose_x(const float* __restrict__ x, float* __restrict__ xT) {
    int i = blockIdx.x * blockDim.x + threadIdx.x;   // i = v*64 + b
    if (i >= NVARS * NBATCH) return;
    int v = i >> 6, b = i & 63;
    xT[i] = x[b * NVARS + v];
}

__global__ void k_aexp(const float* __restrict__ embT, const float* __restrict__ xT,
                       float* __restrict__ AexpT) {  // [(k*2048+v)][64]
    int i = blockIdx.x * blockDim.x + threadIdx.x;
    if (i >= KTOT * NBATCH) return;
    int kv = i >> 6, b = i & 63;
    int k = kv >> 11, v = kv & (NVARS - 1);
    AexpT[i] = embT[k * NBATCH + b] * xT[v * NBATCH + b];
}

// ---------------- GCN scalar aggregation ----------------

__global__ void k_node_self(const float* __restrict__ dinv,
                            const float* __restrict__ src,   // [n][b]
                            float* __restrict__ dst) {       // [n][b]
    int i = blockIdx.x * blockDim.x + threadIdx.x;
    if (i >= NVARS * NBATCH) return;
    float d = dinv[i >> 6];
    dst[i] = d * d * src[i];
}

__global__ void k_edge_agg(const int* __restrict__ ei,
                           const float* __restrict__ dinv,
                           const float* __restrict__ src,    // [n][b]
                           float* __restrict__ dst) {        // [n][b]
    int idx = blockIdx.x * blockDim.x + threadIdx.x;  // e*64 + b
    if (idx >= NEDGE * NBATCH) return;
    int e = idx >> 6, b = idx & 63;
    int r = ei[e];
    int c = ei[NEDGE + e];
    float w = dinv[r] * dinv[c];
    atomicAdd(&dst[c * NBATCH + b], w * src[r * NBATCH + b]);
}

__global__ void k_mlp(const float* __restrict__ ST,
                      const float* __restrict__ W1, const float* __restrict__ b1,
                      const float* __restrict__ W2,
                      float* __restrict__ g) {
    int i = blockIdx.x * blockDim.x + threadIdx.x;
    if (i >= NVARS * NBATCH) return;
    float s = ST[i];
    float acc = 0.0f;
#pragma unroll
    for (int f = 0; f < NHID; ++f)
        acc += tanhf(W1[f] * s + b1[f]) * W2[f];
    g[i] = acc;
}

// ---------------- WMMA GEMM: D[u,b] = sum_kv L'[u,kv] * AexpT[kv,b] ----------------

__global__ __launch_bounds__(32)
void k_gemm(const float* __restrict__ Lall,     // [5][2048][2048]
            const float* __restrict__ AexpT,    // [10240][64]
            float* __restrict__ partial) {      // [NCHUNK][2048][64]
    int wave  = blockIdx.x;
    int ut    = wave % UTILES;
    int chunk = wave / UTILES;
    int kb = chunk / CHUNKS_PER_BASIS;                 // which basis
    int v0 = (chunk % CHUNKS_PER_BASIS) * KCHUNK;      // v offset within basis

    int lane = threadIdx.x & 31;
    int l15  = lane & 15;
    int hi   = lane >> 4;      // 0 for lanes 0-15, 1 for lanes 16-31
    int u0   = ut * 16;

    // A: lane reads row (u0+l15) of basis kb; per K-step lane holds K = vb+2*hi, vb+2*hi+1
    const float* Arow = Lall + (size_t)kb * NVARS * NVARS
                             + (size_t)(u0 + l15) * NVARS + v0 + 2 * hi;
    // B: AexpT rows starting at K = kb*2048 + v0
    const float* Bbase = AexpT + (size_t)(kb * NVARS + v0) * NBATCH;

    v8f c0 = {}, c1 = {}, c2 = {}, c3 = {};

    int r0 = 2 * hi;       // B-frag vgpr0 holds rows K0 (lanes 0-15) / K2 (lanes 16-31)
    int r1 = 2 * hi + 1;   // B-frag vgpr1 holds rows K1 / K3

#pragma unroll 2
    for (int s = 0; s < KCHUNK / 4; ++s) {
        int vb = 4 * s;
        v2f a = *reinterpret_cast<const v2f*>(Arow + vb);   // 8B aligned

        const float* Bp = Bbase + (size_t)vb * NBATCH;
        v2f b0, b1, b2, b3;
        b0.x = Bp[r0 * NBATCH +  0 + l15];  b0.y = Bp[r1 * NBATCH +  0 + l15];
        b1.x = Bp[r0 * NBATCH + 16 + l15];  b1.y = Bp[r1 * NBATCH + 16 + l15];
        b2.x = Bp[r0 * NBATCH + 32 + l15];  b2.y = Bp[r1 * NBATCH + 32 + l15];
        b3.x = Bp[r0 * NBATCH + 48 + l15];  b3.y = Bp[r1 * NBATCH + 48 + l15];

        c0 = __builtin_amdgcn_wmma_f32_16x16x4_f32(false, a, false, b0, (short)0, c0, false, false);
        c1 = __builtin_amdgcn_wmma_f32_16x16x4_f32(false, a, false, b1, (short)0, c1, false, false);
        c2 = __builtin_amdgcn_wmma_f32_16x16x4_f32(false, a, false, b2, (short)0, c2, false, false);
        c3 = __builtin_amdgcn_wmma_f32_16x16x4_f32(false, a, false, b3, (short)0, c3, false, false);
    }

    // C/D layout: vgpr j -> M = u0 + j (lanes 0-15) or u0 + j + 8 (lanes 16-31), N = b tile + l15
    float* outp = partial + (size_t)chunk * (NVARS * NBATCH);
#pragma unroll
    for (int j = 0; j < 8; ++j) {
        int urow = u0 + j + 8 * hi;
        outp[urow * NBATCH +  0 + l15] = c0[j];
        outp[urow * NBATCH + 16 + l15] = c1[j];
        outp[urow * NBATCH + 32 + l15] = c2[j];
        outp[urow * NBATCH + 48 + l15] = c3[j];
    }
}

// ---------------- final combine (deterministic chunk reduction) ----------------

__global__ void k_final(const float* __restrict__ partial,
                        const float* __restrict__ GT,     // [n][b]
                        const float* __restrict__ alpha,  // [b]
                        const float* __restrict__ b2,
                        float* __restrict__ out) {        // [b][n]
    int i = blockIdx.x * blockDim.x + threadIdx.x;   // i = u*64 + b
    if (i >= NVARS * NBATCH) return;
    int u = i >> 6, b = i & 63;
    float s = 0.0f;
#pragma unroll
    for (int c = 0; c < NCHUNK; ++c)
        s += partial[(size_t)c * (NVARS * NBATCH) + i];
    out[b * NVARS + u] = s + alpha[b] * (GT[i] + b2[0]);
}

// ---------------- host launcher ----------------

extern "C" void kernel_launch(void* const* d_in, const int* in_sizes, int n_in,
                              void* d_out, int out_size, void* d_ws, size_t ws_size,
                              hipStream_t stream) {
    const float* x       = (const float*)d_in[0];
    const float* t_years = (const float*)d_in[1];
    const int*   ei      = (const int*)d_in[2];
    const float* Lb      = (const float*)d_in[3];
    const float* W1      = (const float*)d_in[4];
    const float* b1      = (const float*)d_in[5];
    const float* W2      = (const float*)d_in[6];
    const float* b2      = (const float*)d_in[7];
    const float* alpha_w = (const float*)d_in[8];
    float* out = (float*)d_out;

    // workspace carving (floats)
    float* w = (float*)d_ws;
    float* deg   = w;                 w += NVARS;            // 2048
    float* dinv  = w;                 w += NVARS;            // 2048
    float* embT  = w;                 w += NBASIS * NBATCH;  // 320
    float* alpha = w;                 w += NBATCH;           // 64
    float* xT    = w;                 w += NVARS * NBATCH;   // 131072
    float* ST    = w;                 w += NVARS * NBATCH;
    float* g     = w;                 w += NVARS * NBATCH;
    float* GT    = w;                 w += NVARS * NBATCH;
    float* AexpT = w;                 w += KTOT * NBATCH;    // 655360
    float* part  = w;                 w += (size_t)NCHUNK * NVARS * NBATCH; // 2621440

    const int T = 256;
    // prep
    k_emb<<<1, 64, 0, stream>>>(t_years, alpha_w, embT, alpha);
    k_deg_init<<<NVARS / T, T, 0, stream>>>(deg);
    k_deg_acc<<<NEDGE / T, T, 0, stream>>>(ei, deg);
    k_dinv<<<NVARS / T, T, 0, stream>>>(deg, dinv);
    k_transpose_x<<<NVARS * NBATCH / T, T, 0, stream>>>(x, xT);
    k_aexp<<<KTOT * NBATCH / T, T, 0, stream>>>(embT, xT, AexpT);

    // GCN layer 1 aggregation: S
    k_node_self<<<NVARS * NBATCH / T, T, 0, stream>>>(dinv, xT, ST);
    k_edge_agg<<<NEDGE * NBATCH / T, T, 0, stream>>>(ei, dinv, xT, ST);
    // pointwise MLP (tanh hidden, dot with W2)
    k_mlp<<<NVARS * NBATCH / T, T, 0, stream>>>(ST, W1, b1, W2, g);
    // GCN layer 2 aggregation: GT
    k_node_self<<<NVARS * NBATCH / T, T, 0, stream>>>(dinv, g, GT);
    k_edge_agg<<<NEDGE * NBATCH / T, T, 0, stream>>>(ei, dinv, g, GT);

    // big seasonal-operator GEMM on the matrix pipe
    k_gemm<<<UTILES * NCHUNK, 32, 0, stream>>>(Lb, AexpT, part);

    // deterministic reduction + epilogue
    k_final<<<NVARS * NBATCH / T, T, 0, stream>>>(part, GT, alpha, b2, out);
}